// IELT_DINOv2_32255204393650
// MI455X (gfx1250) — compile-verified
//
#include <hip/hip_runtime.h>

// Problem constants (from reference setup_inputs)
#define B_    8
#define NH    12
#define NTOK  1025
#define P     1024     // patches = NTOK-1, 32x32 grid
#define C     768
#define VOTE  24
#define NTILE (C / 16) // 48 channel tiles

typedef __attribute__((ext_vector_type(2))) float v2f;
typedef __attribute__((ext_vector_type(8))) float v8f;

// ---------------------------------------------------------------------------
// Kernel 0: zero the vote-count workspace (atomics need clean state each call)
// ---------------------------------------------------------------------------
__global__ void zero_ws_kernel(float* __restrict__ p, int n) {
  int i = blockIdx.x * blockDim.x + threadIdx.x;
  if (i < n) p[i] = 0.0f;
}

// ---------------------------------------------------------------------------
// Kernel 1: per (batch, head) top-24 of score = weights[b,h,0,1:], vote count.
// One 256-thread block per (b,h). Iterative argmax with min-index tie-break
// (matches jax.lax.top_k tie semantics).
// ---------------------------------------------------------------------------
__global__ void topk_count_kernel(const float* __restrict__ weights,
                                  float* __restrict__ cnt_raw) {
  __shared__ float vals[P];
  __shared__ float rv[256];
  __shared__ int   ri[256];
  const int bh  = blockIdx.x;          // b*NH + h
  const int b   = bh / NH;
  const int tid = threadIdx.x;

  const float* score = weights + (size_t)bh * NTOK * NTOK + 1; // row 0, col 1..
  for (int p = tid; p < P; p += 256) vals[p] = score[p];
  __syncthreads();

  for (int it = 0; it < VOTE; ++it) {
    float bv = -__builtin_inff(); int bi = P;
    for (int p = tid; p < P; p += 256) {
      float v = vals[p];
      if (v > bv) { bv = v; bi = p; }     // strict > keeps smallest index
    }
    rv[tid] = bv; ri[tid] = bi;
    __syncthreads();
    for (int s = 128; s > 0; s >>= 1) {
      if (tid < s) {
        float ov = rv[tid + s]; int oi = ri[tid + s];
        if (ov > rv[tid] || (ov == rv[tid] && oi < ri[tid])) { rv[tid] = ov; ri[tid] = oi; }
      }
      __syncthreads();
    }
    if (tid == 0) {
      int w = ri[0];
      atomicAdd(&cnt_raw[(size_t)b * P + w], 1.0f);  // votes accumulate across heads
      vals[w] = -__builtin_inff();
    }
    __syncthreads();
  }
}

// ---------------------------------------------------------------------------
// Kernel 2: _enhance_local — 32x32 image, 3x3 kernel [[1,2,1],[2,4,2],[1,2,1]]
// at dilations {1,2,4}, zero padded SAME, max over dilations.
// One 1024-thread block per batch; image staged in LDS.
// ---------------------------------------------------------------------------
__global__ void enhance_kernel(const float* __restrict__ cnt_raw,
                               float* __restrict__ cnt_enh,
                               float* __restrict__ out_count) {
  __shared__ float img[P];
  const int b   = blockIdx.x;
  const int tid = threadIdx.x;         // 0..1023
  img[tid] = cnt_raw[(size_t)b * P + tid];
  __syncthreads();

  const int y = tid >> 5, x = tid & 31;
  const float kw[3][3] = {{1.f,2.f,1.f},{2.f,4.f,2.f},{1.f,2.f,1.f}};
  float m = -__builtin_inff();
#pragma unroll
  for (int di = 0; di < 3; ++di) {
    const int d = 1 << di;             // 1, 2, 4
    float acc = 0.0f;
#pragma unroll
    for (int dy = -1; dy <= 1; ++dy)
#pragma unroll
      for (int dx = -1; dx <= 1; ++dx) {
        int yy = y + dy * d, xx = x + dx * d;
        if (yy >= 0 && yy < 32 && xx >= 0 && xx < 32)
          acc += kw[dy + 1][dx + 1] * img[yy * 32 + xx];
      }
    m = fmaxf(m, acc);
  }
  cnt_enh[(size_t)b * P + tid]   = m;
  out_count[(size_t)b * P + tid] = m;  // second output of the reference tuple
}

// ---------------------------------------------------------------------------
// Kernel 3a: per batch — top-SEL of enhanced count (stable, min-index ties ==
// argsort(-count)). Writes: masked count vector (selected entries zeroed ==
// reference's unselect gather), 1/(sum+1e-6) scale, and the SEL forwarded
// tokens (reference gathers hidden_states at patch indices with NO +1 shift).
// ---------------------------------------------------------------------------
__global__ void select_kernel(const float* __restrict__ hidden,
                              const float* __restrict__ cnt_enh,
                              float* __restrict__ cnt_masked,
                              float* __restrict__ scale_ws,
                              float* __restrict__ out_tokens,
                              int selp1) {
  __shared__ float s_cnt[P];
  __shared__ float rv[512];
  __shared__ int   ri[512];
  __shared__ int   s_sel[64];

  const int b   = blockIdx.x;
  const int tid = threadIdx.x;         // 0..511
  int SEL = selp1 - 1;
  if (SEL > 64) SEL = 64;

  for (int p = tid; p < P; p += 512) s_cnt[p] = cnt_enh[(size_t)b * P + p];
  __syncthreads();

  // --- top-SEL (iterative argmax, min-index tie-break) ---
  for (int it = 0; it < SEL; ++it) {
    float bv = -__builtin_inff(); int bi = P;
    for (int p = tid; p < P; p += 512) {
      float v = s_cnt[p];
      if (v > bv) { bv = v; bi = p; }
    }
    rv[tid] = bv; ri[tid] = bi;
    __syncthreads();
    for (int s = 256; s > 0; s >>= 1) {
      if (tid < s) {
        float ov = rv[tid + s]; int oi = ri[tid + s];
        if (ov > rv[tid] || (ov == rv[tid] && oi < ri[tid])) { rv[tid] = ov; ri[tid] = oi; }
      }
      __syncthreads();
    }
    if (tid == 0) { s_sel[it] = ri[0]; s_cnt[ri[0]] = -1.0f; }  // counts >= 0: -1 never re-picked
    __syncthreads();
  }
  if (tid < SEL) s_cnt[s_sel[tid]] = 0.0f;   // exclude selected from the merge
  __syncthreads();

  // --- masked counts out + denominator ---
  {
    float acc = 0.0f;
    for (int p = tid; p < P; p += 512) {
      float v = s_cnt[p];
      cnt_masked[(size_t)b * P + p] = v;
      acc += v;
    }
    rv[tid] = acc;
    __syncthreads();
    for (int s = 256; s > 0; s >>= 1) {
      if (tid < s) rv[tid] += rv[tid + s];
      __syncthreads();
    }
    if (tid == 0) scale_ws[b] = 1.0f / (rv[0] + 1e-6f);
  }

  // --- forwarded tokens (small: SEL*768 per batch) ---
  for (int i = tid; i < SEL * C; i += 512) {
    int j = i / C, c = i % C;
    out_tokens[((size_t)b * selp1 + j) * C + c] =
        hidden[((size_t)b * NTOK + s_sel[j]) * C + c];
  }
}

// ---------------------------------------------------------------------------
// Kernel 3b: merged token via V_WMMA_F32_16X16X4_F32.
// One block per (batch, 16-channel tile) -> B*48 = 384 blocks so the 25 MB
// hidden_states read is spread across the whole chip (8 blocks could never
// feed 23.3 TB/s). 2 waves per block split K=1024 in halves (shorter WMMA
// dependency chain, 2x memory-level parallelism); partials meet in LDS.
// A operand = masked count broadcast into all 16 M-rows (row 0 of D is the
// desired dot product); B operand = 4x16 f32 tiles of hidden_states.
// EXEC is all-1s at every WMMA (uniform control flow, full waves).
// ---------------------------------------------------------------------------
__global__ void merge_wmma_kernel(const float* __restrict__ hidden,
                                  const float* __restrict__ cnt_masked,
                                  const float* __restrict__ scale_ws,
                                  float* __restrict__ out_tokens,
                                  int selp1) {
  __shared__ float s_cnt[P];
  __shared__ float s_part[2][16];
  const int b    = blockIdx.x / NTILE;
  const int tile = blockIdx.x % NTILE;
  const int tid  = threadIdx.x;        // 0..63 (2 waves)

  for (int p = tid; p < P; p += 64) s_cnt[p] = cnt_masked[(size_t)b * P + p];
  __syncthreads();

  const int wave = tid >> 5, lane = tid & 31;
  const int half = lane >> 4, col = lane & 15;   // A: lanes 0-15 hold K0/K1, 16-31 hold K2/K3
  const float* hb = hidden + (size_t)b * NTOK * C + C + tile * 16 + col; // token row 1 == patch 0

  v8f acc = {};
  const int k0 = wave * (P / 2), k1 = k0 + P / 2;
  for (int k = k0; k < k1; k += 4) {
    v2f a, bm;
    a.x  = s_cnt[k + 2 * half];                 // K = 0/2
    a.y  = s_cnt[k + 2 * half + 1];             // K = 1/3
    bm.x = hb[(size_t)(k + half) * C];          // B rows K0/K1 across lane halves
    bm.y = hb[(size_t)(k + 2 + half) * C];      // B rows K2/K3
    acc  = __builtin_amdgcn_wmma_f32_16x16x4_f32(
               /*neg_a=*/false, a, /*neg_b=*/false, bm,
               /*c_mod=*/(short)0, acc, /*reuse_a=*/false, /*reuse_b=*/false);
  }
  // D layout: VGPR0, lanes 0-15 = row M=0 (our valid row), N = lane
  if (half == 0) s_part[wave][col] = acc[0];
  __syncthreads();
  if (tid < 16) {
    float num = s_part[0][tid] + s_part[1][tid];
    out_tokens[((size_t)b * selp1 + (selp1 - 1)) * C + tile * 16 + tid] =
        num * scale_ws[b];
  }
}

// ---------------------------------------------------------------------------
extern "C" void kernel_launch(void* const* d_in, const int* in_sizes, int n_in,
                              void* d_out, int out_size, void* d_ws, size_t ws_size,
                              hipStream_t stream) {
  const float* weights = (const float*)d_in[0];   // (8,12,1025,1025) f32
  const float* hidden  = (const float*)d_in[1];   // (8,1025,768) f32

  float* cnt_raw    = (float*)d_ws;               // B*P floats
  float* cnt_enh    = cnt_raw + B_ * P;           // B*P floats
  float* cnt_masked = cnt_enh + B_ * P;           // B*P floats
  float* scale_ws   = cnt_masked + B_ * P;        // B floats

  // out = out_tokens (B,(SEL+1),C) ++ count (B,P); recover SEL+1 from out_size
  int selp1 = (out_size - B_ * P) / (B_ * C);
  float* out_tokens = (float*)d_out;
  float* out_count  = out_tokens + (size_t)B_ * selp1 * C;
  (void)in_sizes; (void)n_in; (void)ws_size;

  zero_ws_kernel<<<(B_ * P + 1023) / 1024, 1024, 0, stream>>>(cnt_raw, B_ * P);
  topk_count_kernel<<<B_ * NH, 256, 0, stream>>>(weights, cnt_raw);
  enhance_kernel<<<B_, 1024, 0, stream>>>(cnt_raw, cnt_enh, out_count);
  select_kernel<<<B_, 512, 0, stream>>>(hidden, cnt_enh, cnt_masked, scale_ws,
                                        out_tokens, selp1);
  merge_wmma_kernel<<<B_ * NTILE, 64, 0, stream>>>(hidden, cnt_masked, scale_ws,
                                                   out_tokens, selp1);
}